// Marginal_41609643163874
// MI455X (gfx1250) — compile-verified
//
#include <hip/hip_runtime.h>
#include <hip/hip_bf16.h>
#include <math.h>

// CDNA5 wave32 WMMA fragment types (V_WMMA_F32_16X16X4_F32: A/B = 2 VGPRs, C/D = 8)
typedef float v2f __attribute__((ext_vector_type(2)));
typedef float v8f __attribute__((ext_vector_type(8)));

#define KCOLS 16384   // (512/4)*(512/4)
#define NROWS 64
#define INV_H 50.0f   // 1/BANDWIDTH

#if defined(__has_builtin)
#if __has_builtin(__builtin_amdgcn_ballot_w32)
#define BALLOT32(p) __builtin_amdgcn_ballot_w32(p)
#endif
#endif
#ifndef BALLOT32
#define BALLOT32(p) ((unsigned int)__ballot(p))
#endif

// One wave (32 lanes) per column j; 8 waves per 256-thread block.
// Computes lk^T (b rows, a cols) via 16 f32 WMMAs so the b-reduction is in-lane,
// then writes partial[j] = sum_a |ratio(a,j) - f(a,j)|.
__global__ void __launch_bounds__(256)
ece_kde_wmma_kernel(const float* __restrict__ inp,
                    const float* __restrict__ tgt,
                    float* __restrict__ partial) {
  __shared__ float sF [8][NROWS];
  __shared__ float sLF[8][NROWS];
  __shared__ float sL1[8][NROWS];
  __shared__ float sC [8][NROWS];

  const int lane = threadIdx.x & 31;
  const int w    = threadIdx.x >> 5;
  const int j    = blockIdx.x * 8 + w;
  const int yi   = j >> 7;
  const int xi   = j & 127;
  const size_t off = (size_t)yi * (4 * 512) + (size_t)xi * 4;

  const float lg52 = lgammaf(52.0f);   // gammaln(pa+qa); pa+qa == 1/h + 2 == 52

  // Stage this column's 64 rows (f, log f, log1p(-f), log Beta) + ballot the y mask.
  unsigned int ym[2];
  #pragma unroll
  for (int rep = 0; rep < 2; ++rep) {
    const int r = lane + rep * 32;
    const size_t idx = (size_t)r * (512 * 512) + off;
    const float f = inp[idx];
    const float y = tgt[idx];
    sF [w][r] = f;
    sLF[w][r] = logf(f);
    sL1[w][r] = log1pf(-f);
    sC [w][r] = lgammaf(f * INV_H + 1.0f) + lgammaf((1.0f - f) * INV_H + 1.0f) - lg52;
    ym[rep]   = BALLOT32(y > 0.0f);    // bit r%32 = y[r] for rows rep*32..rep*32+31
  }
  __syncthreads();

  const int h  = lane >> 4;   // lane half
  const int n  = lane & 15;   // tile column N  (== the 'a' index this lane owns)
  const int rr = 8 * h;       // row offset of this half within a tile

  // Per-lane packed y bits: bit (u*8+v) = y[b] for b = u*16 + v + rr
  unsigned int ybits = 0;
  #pragma unroll
  for (int u = 0; u < 4; ++u) {
    const unsigned int word = (u < 2) ? ym[0] : ym[1];
    ybits |= ((word >> ((u & 1) * 16 + rr)) & 0xFFu) << (u * 8);
  }

  // A-fragments (16x4, M = b): lanes0-15 -> K0,K1 ; lanes16-31 -> K2,K3
  //   X'[b] = [f[b]/h, (1-f[b])/h, -logB[b], 0]
  v2f afrag[4];
  #pragma unroll
  for (int u = 0; u < 4; ++u) {
    const float fb = sF[w][u * 16 + n];
    const float cb = sC[w][u * 16 + n];
    afrag[u].x = (h == 0) ? fb * INV_H          : -cb;
    afrag[u].y = (h == 0) ? (1.0f - fb) * INV_H : 0.0f;
  }

  float acc = 0.0f;

  #pragma unroll
  for (int t = 0; t < 4; ++t) {
    // B-fragment (4x16, N = a): K = vgpr + 2*half
    //   Y'[:,a] = [log f[a], log1p(-f[a]), 1, 0]^T
    v2f bfrag;
    bfrag.x = (h == 0) ? sLF[w][t * 16 + n] : 1.0f;
    bfrag.y = (h == 0) ? sL1[w][t * 16 + n] : 0.0f;

    const v8f z = {0.f, 0.f, 0.f, 0.f, 0.f, 0.f, 0.f, 0.f};
    v8f D[4];   // lk^T tiles: rows b = u*16 + v + rr, col a = t*16 + n
    #pragma unroll
    for (int u = 0; u < 4; ++u) {
      D[u] = __builtin_amdgcn_wmma_f32_16x16x4_f32(
          /*neg_a=*/false, afrag[u], /*neg_b=*/false, bfrag,
          /*c_mod=*/(short)0, z, /*reuse_a=*/false, /*reuse_b=*/false);
    }

    // In-lane over 32 b's, then one cross-half merge per quantity.
    float m = -__builtin_inff();
    #pragma unroll
    for (int u = 0; u < 4; ++u) {
      #pragma unroll
      for (int v = 0; v < 8; ++v) {
        float val = D[u][v];
        if (u == t && n == v + rr) val = -__builtin_inff();  // leave-one-out diag
        D[u][v] = val;
        m = fmaxf(m, val);
      }
    }
    m = fmaxf(m, __shfl_xor(m, 16, 32));   // shared max over all 64 b (cancels num/den)

    float den = 0.0f, num = 0.0f;
    #pragma unroll
    for (int u = 0; u < 4; ++u) {
      #pragma unroll
      for (int v = 0; v < 8; ++v) {
        const float e = __expf(D[u][v] - m);   // masked entries -> exp(-inf) = 0
        den += e;
        num += ((ybits >> (u * 8 + v)) & 1u) ? e : 0.0f;
      }
    }
    den += __shfl_xor(den, 16, 32);
    num += __shfl_xor(num, 16, 32);

    const float ratio = num / den;                    // E[y|f] KDE estimate
    const float err   = fabsf(ratio - sF[w][t * 16 + n]);  // P = 1
    if (h == 0) acc += err;                           // half 0 owns row a = t*16+n
  }

  // Wave reduction of the 16 per-a partials (half1 lanes carry 0).
  #pragma unroll
  for (int s = 1; s < 32; s <<= 1) acc += __shfl_xor(acc, s, 32);
  if (lane == 0) partial[j] = acc;
}

// Deterministic fixed-order reduction of the 16384 per-column partials.
__global__ void __launch_bounds__(256)
ece_kde_reduce_kernel(const float* __restrict__ partial, float* __restrict__ out) {
  __shared__ float s[256];
  float a = 0.0f;
  for (int i = threadIdx.x; i < KCOLS; i += 256) a += partial[i];
  s[threadIdx.x] = a;
  __syncthreads();
  #pragma unroll
  for (int st = 128; st > 0; st >>= 1) {
    if ((int)threadIdx.x < st) s[threadIdx.x] += s[threadIdx.x + st];
    __syncthreads();
  }
  if (threadIdx.x == 0) out[0] = s[0] * (1.0f / (float)NROWS);  // mean over axis 0
}

extern "C" void kernel_launch(void* const* d_in, const int* in_sizes, int n_in,
                              void* d_out, int out_size, void* d_ws, size_t ws_size,
                              hipStream_t stream) {
  const float* inp = (const float*)d_in[0];   // (64,1,512,512) f32
  const float* tgt = (const float*)d_in[1];   // (64,1,512,512) f32
  float* out     = (float*)d_out;             // scalar f32
  float* partial = (float*)d_ws;              // 16384 floats of scratch

  ece_kde_wmma_kernel<<<KCOLS / 8, 256, 0, stream>>>(inp, tgt, partial);
  ece_kde_reduce_kernel<<<1, 256, 0, stream>>>(partial, out);
}